// AllPoleDigitalFilter_55009941127447
// MI455X (gfx1250) — compile-verified
//
#include <hip/hip_runtime.h>
#include <hip/hip_bf16.h>

#define ORDER   30
#define NFRAMES 200
#define NB      64
#define PERIOD  80
#define TLEN    (NFRAMES * PERIOD)
#define SDIM    32                  // padded state: 30 taps + homogeneous-1 + pad
#define GRP     8                   // chunks per group
#define NGRP    (NFRAMES / GRP)     // 25
#define LSTRIDE 33                  // padded LDS stride (bank-conflict relief)

typedef __attribute__((ext_vector_type(2))) float v2f;
typedef __attribute__((ext_vector_type(8))) float v8f;

// ---------------------------------------------------------------------------
// K1: build per-chunk 32x32 homogeneous transition matrices.
// One wave per chunk; lane j simulates column j (lane 30 = input column).
// ---------------------------------------------------------------------------
__global__ __launch_bounds__(32) void k1_build(const float* __restrict__ x,
                                               const float* __restrict__ a,
                                               float* __restrict__ Mws) {
    const int chunk = blockIdx.x;              // b*NFRAMES + c
    const int b = chunk / NFRAMES;
    const int c = chunk - b * NFRAMES;
    const int j = threadIdx.x;                 // column index (lane)

    const int cn = (c + 1 < NFRAMES) ? (c + 1) : (NFRAMES - 1);
    const float* A0 = a + ((size_t)b * NFRAMES + c)  * (ORDER + 1);
    const float* A1 = a + ((size_t)b * NFRAMES + cn) * (ORDER + 1);

    float a0[ORDER], da[ORDER];
#pragma unroll
    for (int m = 0; m < ORDER; ++m) { a0[m] = A0[m + 1]; da[m] = A1[m + 1] - a0[m]; }
    const float K0 = A0[0], dK = A1[0] - K0;

    float hist[ORDER];
#pragma unroll
    for (int m = 0; m < ORDER; ++m) hist[m] = (j == m) ? 1.0f : 0.0f;
    const float inMask = (j == ORDER) ? 1.0f : 0.0f;   // lane 30: input-driven column
    const float* xp = x + (size_t)b * TLEN + (size_t)c * PERIOD;

    for (int t = 0; t < PERIOD; ++t) {
        const float f = (float)t * (1.0f / PERIOD);
        float acc0 = 0.0f, acc1 = 0.0f;
#pragma unroll
        for (int m = 0; m < ORDER; ++m) {
            acc0 = fmaf(a0[m], hist[m], acc0);
            acc1 = fmaf(da[m], hist[m], acc1);
        }
        const float g = fmaf(f, dK, K0);
        const float y = inMask * (g * xp[t]) - fmaf(f, acc1, acc0);
#pragma unroll
        for (int m = ORDER - 1; m > 0; --m) hist[m] = hist[m - 1];
        hist[0] = y;
    }

    float* Mo = Mws + (size_t)chunk * (SDIM * SDIM);
#pragma unroll
    for (int r = 0; r < ORDER; ++r) Mo[r * SDIM + j] = hist[r]; // cols 31 are 0 naturally
    Mo[30 * SDIM + j] = (j == 30) ? 1.0f : 0.0f;                // preserve homogeneous 1
    Mo[31 * SDIM + j] = 0.0f;
}

// ---------------------------------------------------------------------------
// 32x32 f32 matmul O = L * R via V_WMMA_F32_16X16X4_F32 (LDS staged, stride 33)
// ---------------------------------------------------------------------------
__device__ __forceinline__ void matmul32_lds(const float* __restrict__ L,
                                             const float* __restrict__ R,
                                             float* __restrict__ O, int lane) {
    const int half = lane >> 4;   // 0: lanes 0-15, 1: lanes 16-31
    const int lm   = lane & 15;

    v8f d00 = {}, d01 = {}, d10 = {}, d11 = {};
#pragma unroll
    for (int ks = 0; ks < 8; ++ks) {            // K slices of 4
        const int kb = ks * 4 + 2 * half;
        v2f aA, aB, bA, bB;
        // A fragments (16x4): lane holds A[m][kb], A[m][kb+1]
        aA.x = L[(lm)      * LSTRIDE + kb];  aA.y = L[(lm)      * LSTRIDE + kb + 1];
        aB.x = L[(16 + lm) * LSTRIDE + kb];  aB.y = L[(16 + lm) * LSTRIDE + kb + 1];
        // B fragments (4x16): lane holds B[kb][col], B[kb+1][col]
        bA.x = R[(kb)     * LSTRIDE + lm];       bA.y = R[(kb + 1) * LSTRIDE + lm];
        bB.x = R[(kb)     * LSTRIDE + 16 + lm];  bB.y = R[(kb + 1) * LSTRIDE + 16 + lm];

        d00 = __builtin_amdgcn_wmma_f32_16x16x4_f32(false, aA, false, bA, (short)0, d00, false, false);
        d01 = __builtin_amdgcn_wmma_f32_16x16x4_f32(false, aA, false, bB, (short)0, d01, false, false);
        d10 = __builtin_amdgcn_wmma_f32_16x16x4_f32(false, aB, false, bA, (short)0, d10, false, false);
        d11 = __builtin_amdgcn_wmma_f32_16x16x4_f32(false, aB, false, bB, (short)0, d11, false, false);
    }
    __syncthreads();   // O may alias R: all reads done before overwrite
#pragma unroll
    for (int r = 0; r < 8; ++r) {
        const int row0 = r + 8 * half;          // C/D layout: VGPR r -> row r / r+8
        O[(row0)      * LSTRIDE + lm]      = d00[r];
        O[(row0)      * LSTRIDE + 16 + lm] = d01[r];
        O[(16 + row0) * LSTRIDE + lm]      = d10[r];
        O[(16 + row0) * LSTRIDE + 16 + lm] = d11[r];
    }
}

// ---------------------------------------------------------------------------
// K2: compose 8 chunk matrices per group: G = M[7]*M[6]*...*M[0]  (WMMA)
// ---------------------------------------------------------------------------
__global__ __launch_bounds__(32) void k2_compose(const float* __restrict__ Mws,
                                                 float* __restrict__ Gws) {
    __shared__ float cur[SDIM * LSTRIDE];
    __shared__ float lft[SDIM * LSTRIDE];
    const int grp  = blockIdx.x;               // b*NGRP + g
    const int lane = threadIdx.x;
    const int b = grp / NGRP, g = grp - b * NGRP;
    const int chunk0 = b * NFRAMES + g * GRP;

    for (int e = lane; e < SDIM * SDIM; e += 32) {
        const int r = e >> 5, cc = e & 31;
        cur[r * LSTRIDE + cc] = Mws[(size_t)chunk0 * (SDIM * SDIM) + e];
    }
    __syncthreads();
    for (int i = 1; i < GRP; ++i) {
        for (int e = lane; e < SDIM * SDIM; e += 32) {
            const int r = e >> 5, cc = e & 31;
            lft[r * LSTRIDE + cc] = Mws[((size_t)chunk0 + i) * (SDIM * SDIM) + e];
        }
        __syncthreads();
        matmul32_lds(lft, cur, cur, lane);     // cur = M[i] * cur
        __syncthreads();
    }
    for (int e = lane; e < SDIM * SDIM; e += 32) {
        const int r = e >> 5, cc = e & 31;
        Gws[(size_t)grp * (SDIM * SDIM) + e] = cur[r * LSTRIDE + cc];
    }
}

// ---------------------------------------------------------------------------
// K3: sequential scan over 25 group composites per batch (exclusive starts)
// ---------------------------------------------------------------------------
__global__ __launch_bounds__(32) void k3_scan_groups(const float* __restrict__ Gws,
                                                     float* __restrict__ hgrp) {
    __shared__ float sh[SDIM];
    const int b = blockIdx.x;
    const int r = threadIdx.x;
    sh[r] = (r == 30) ? 1.0f : 0.0f;           // homogeneous initial state
    __syncthreads();
    for (int g = 0; g < NGRP; ++g) {
        hgrp[((size_t)b * NGRP + g) * SDIM + r] = sh[r];
        const float* G = Gws + ((size_t)b * NGRP + g) * (SDIM * SDIM) + r * SDIM;
        float acc = 0.0f;
#pragma unroll
        for (int jj = 0; jj < SDIM; ++jj) acc = fmaf(G[jj], sh[jj], acc);
        __syncthreads();
        sh[r] = acc;
        __syncthreads();
    }
}

// ---------------------------------------------------------------------------
// K4: within each group, matvec-walk the 8 chunks to emit per-chunk starts
// ---------------------------------------------------------------------------
__global__ __launch_bounds__(32) void k4_distribute(const float* __restrict__ Mws,
                                                    const float* __restrict__ hgrp,
                                                    float* __restrict__ hstart) {
    __shared__ float sh[SDIM];
    const int grp = blockIdx.x;
    const int b = grp / NGRP, g = grp - b * NGRP;
    const int r = threadIdx.x;
    sh[r] = hgrp[(size_t)grp * SDIM + r];
    __syncthreads();
    for (int i = 0; i < GRP; ++i) {
        const size_t chunk = (size_t)b * NFRAMES + g * GRP + i;
        hstart[chunk * SDIM + r] = sh[r];
        const float* M = Mws + chunk * (SDIM * SDIM) + r * SDIM;
        float acc = 0.0f;
#pragma unroll
        for (int jj = 0; jj < SDIM; ++jj) acc = fmaf(M[jj], sh[jj], acc);
        __syncthreads();
        sh[r] = acc;
        __syncthreads();
    }
}

// ---------------------------------------------------------------------------
// K5: replay each chunk from its true start history; thread per chunk
// ---------------------------------------------------------------------------
__global__ __launch_bounds__(256) void k5_replay(const float* __restrict__ x,
                                                 const float* __restrict__ a,
                                                 const float* __restrict__ hstart,
                                                 float* __restrict__ y) {
    const int chunk = blockIdx.x * blockDim.x + threadIdx.x;
    if (chunk >= NB * NFRAMES) return;
    const int b = chunk / NFRAMES, c = chunk - b * NFRAMES;

    const int cn = (c + 1 < NFRAMES) ? (c + 1) : (NFRAMES - 1);
    const float* A0 = a + ((size_t)b * NFRAMES + c)  * (ORDER + 1);
    const float* A1 = a + ((size_t)b * NFRAMES + cn) * (ORDER + 1);
    float a0[ORDER], da[ORDER];
#pragma unroll
    for (int m = 0; m < ORDER; ++m) { a0[m] = A0[m + 1]; da[m] = A1[m + 1] - a0[m]; }
    const float K0 = A0[0], dK = A1[0] - K0;

    float hist[ORDER];
    const float* h0 = hstart + (size_t)chunk * SDIM;
#pragma unroll
    for (int m = 0; m < ORDER; ++m) hist[m] = h0[m];

    const float* xp = x + (size_t)b * TLEN + (size_t)c * PERIOD;
    float*       yp = y + (size_t)b * TLEN + (size_t)c * PERIOD;

    for (int t = 0; t < PERIOD; ++t) {
        const float f = (float)t * (1.0f / PERIOD);
        float acc0 = 0.0f, acc1 = 0.0f;
#pragma unroll
        for (int m = 0; m < ORDER; ++m) {
            acc0 = fmaf(a0[m], hist[m], acc0);
            acc1 = fmaf(da[m], hist[m], acc1);
        }
        const float g  = fmaf(f, dK, K0);
        const float yt = g * xp[t] - fmaf(f, acc1, acc0);
        yp[t] = yt;
#pragma unroll
        for (int m = ORDER - 1; m > 0; --m) hist[m] = hist[m - 1];
        hist[0] = yt;
    }
}

// ---------------------------------------------------------------------------
extern "C" void kernel_launch(void* const* d_in, const int* in_sizes, int n_in,
                              void* d_out, int out_size, void* d_ws, size_t ws_size,
                              hipStream_t stream) {
    const float* x = (const float*)d_in[0];    // (B, T) f32
    const float* a = (const float*)d_in[1];    // (B, N, 31) f32
    float* out = (float*)d_out;                // (B, T) f32

    float* Mws    = (float*)d_ws;                                    // 12800 * 1024 f32
    float* Gws    = Mws    + (size_t)NB * NFRAMES * SDIM * SDIM;     //  1600 * 1024 f32
    float* hgrp   = Gws    + (size_t)NB * NGRP   * SDIM * SDIM;      //  1600 *   32 f32
    float* hstart = hgrp   + (size_t)NB * NGRP   * SDIM;             // 12800 *   32 f32

    k1_build      <<<NB * NFRAMES, 32,  0, stream>>>(x, a, Mws);
    k2_compose    <<<NB * NGRP,    32,  0, stream>>>(Mws, Gws);
    k3_scan_groups<<<NB,           32,  0, stream>>>(Gws, hgrp);
    k4_distribute <<<NB * NGRP,    32,  0, stream>>>(Mws, hgrp, hstart);
    k5_replay     <<<(NB * NFRAMES + 255) / 256, 256, 0, stream>>>(x, a, hstart, out);
}